// AdvanceGATModel_20882130993666
// MI455X (gfx1250) — compile-verified
//
#include <hip/hip_runtime.h>

// ---------------- model constants (from reference) ----------------
#define F_IN   128
#define HIDC   32
#define OUTC   4
#define NHEAD  8
#define HC1    (NHEAD * HIDC)   // 256
#define HC2    (NHEAD * OUTC)   // 32
#define NEG    0.2f

typedef __attribute__((ext_vector_type(2))) float v2f;
typedef __attribute__((ext_vector_type(8))) float v8f;

// ---- order-preserving float<->uint encoding for atomicMax on floats ----
__device__ __forceinline__ unsigned enc_f(float f) {
  unsigned u = __float_as_uint(f);
  return (u & 0x80000000u) ? ~u : (u | 0x80000000u);
}
__device__ __forceinline__ float dec_f(unsigned u) {
  return __uint_as_float((u & 0x80000000u) ? (u & 0x7FFFFFFFu) : ~u);
}

__device__ __forceinline__ void get_edge(const int* __restrict__ ei, long long E,
                                         long long e, int& s, int& d) {
  if (e < E) { s = ei[e]; d = ei[E + e]; }
  else       { s = d = (int)(e - E); }   // appended self-loops
}

__device__ __forceinline__ float lrelu(float x) { return x > 0.f ? x : NEG * x; }

// ---------------- init kernels ----------------
__global__ void k_fill_zero(float* __restrict__ p, long long n) {
  long long i = (long long)blockIdx.x * blockDim.x + threadIdx.x;
  if (i < n) p[i] = 0.f;
}
__global__ void k_init_softmax(unsigned* __restrict__ amax, float* __restrict__ denom,
                               long long n) {
  long long i = (long long)blockIdx.x * blockDim.x + threadIdx.x;
  if (i < n) { amax[i] = 0x007FFFFFu /* enc(-inf) */; denom[i] = 0.f; }
}

// ---------------- fp32 WMMA GEMM: Out[M,NC] = X[M,K] @ W[K,NC] ----------------
// One wave computes a 16 x (16*NT) strip: the A fragment (16x4 f32) is loaded
// once per k-step and reused across NT column tiles (V_WMMA_F32_16X16X4_F32).
// A frag layout (ISA 7.12.2): lanes 0-15 -> K=0,1 ; lanes 16-31 -> K=2,3 ; M = lane&15.
// B frag (4x16): same pattern with N = lane&15.
// C/D: VGPR i holds M = i + 8*(lane>=16), N = lane&15.
template<int NT>
__global__ void k_gemm_wmma(const float* __restrict__ X, const float* __restrict__ W,
                            float* __restrict__ Out, int K, int NC,
                            int groupsN, int groupsTotal) {
  int wave = (int)(((long long)blockIdx.x * blockDim.x + threadIdx.x) >> 5);
  int lane = threadIdx.x & 31;
  if (wave >= groupsTotal) return;           // wave-uniform: EXEC all-1s inside

  int tm = wave / groupsN;
  int tg = wave - tm * groupsN;
  int colBase = tg * 16 * NT;
  int row = lane & 15;
  int ks  = (lane >> 4) << 1;                // 0 for lanes 0-15, 2 for lanes 16-31
  const float* xrow = X + (long long)(tm * 16 + row) * K;

  v8f acc[NT];
  #pragma unroll
  for (int t = 0; t < NT; ++t) acc[t] = (v8f){};

  for (int k0 = 0; k0 < K; k0 += 4) {
    v2f a;
    a.x = xrow[k0 + ks];
    a.y = xrow[k0 + ks + 1];
    const float* w0 = W + (long long)(k0 + ks)     * NC + colBase + row;
    const float* w1 = W + (long long)(k0 + ks + 1) * NC + colBase + row;
    #pragma unroll
    for (int t = 0; t < NT; ++t) {
      v2f b;
      b.x = w0[t * 16];
      b.y = w1[t * 16];
      acc[t] = __builtin_amdgcn_wmma_f32_16x16x4_f32(false, a, false, b,
                                                     (short)0, acc[t], false, false);
    }
  }

  int mb   = (lane >> 4) * 8;
  int ccol = lane & 15;
  #pragma unroll
  for (int t = 0; t < NT; ++t)
    #pragma unroll
    for (int i = 0; i < 8; ++i)
      Out[(long long)(tm * 16 + mb + i) * NC + colBase + t * 16 + ccol] = acc[t][i];
}

// ---------------- per-node attention logits ----------------
// as_[n,h] = sum_c h[n,h,c]*a_src[h,c]   (same for ad_)
template<int C>
__global__ void k_alpha(const float* __restrict__ h, const float* __restrict__ a_src,
                        const float* __restrict__ a_dst, float* __restrict__ as_,
                        float* __restrict__ ad_, long long n) {
  long long i = (long long)blockIdx.x * blockDim.x + threadIdx.x;
  if (i >= n * NHEAD) return;
  long long node = i / NHEAD;
  int hd = (int)(i - node * NHEAD);
  const float* hp = h + (node * NHEAD + hd) * C;
  const float* sp = a_src + hd * C;
  const float* dp = a_dst + hd * C;
  float s = 0.f, d = 0.f;
  #pragma unroll
  for (int c = 0; c < C; ++c) { float v = hp[c]; s += v * sp[c]; d += v * dp[c]; }
  as_[i] = s; ad_[i] = d;
}

// ---------------- edge pass 1: segment max over dst ----------------
__global__ void k_edge_max(const int* __restrict__ ei, long long E, long long Ep,
                           const float* __restrict__ as_, const float* __restrict__ ad_,
                           unsigned* __restrict__ amax) {
  long long i = (long long)blockIdx.x * blockDim.x + threadIdx.x;
  if (i >= Ep * NHEAD) return;
  long long e = i / NHEAD;
  int hd = (int)(i - e * NHEAD);
  int s, d; get_edge(ei, E, e, s, d);
  float a = lrelu(as_[(long long)s * NHEAD + hd] + ad_[(long long)d * NHEAD + hd]);
  atomicMax(&amax[(long long)d * NHEAD + hd], enc_f(a));
}

// ---------------- edge pass 2: ex = exp(alpha - amax[dst]); denom += ex ----------------
__global__ void k_edge_exp(const int* __restrict__ ei, long long E, long long Ep,
                           const float* __restrict__ as_, const float* __restrict__ ad_,
                           const unsigned* __restrict__ amax, float* __restrict__ denom,
                           float* __restrict__ ex) {
  long long i = (long long)blockIdx.x * blockDim.x + threadIdx.x;
  if (i >= Ep * NHEAD) return;
  long long e = i / NHEAD;
  int hd = (int)(i - e * NHEAD);
  int s, d; get_edge(ei, E, e, s, d);
  float a = lrelu(as_[(long long)s * NHEAD + hd] + ad_[(long long)d * NHEAD + hd]);
  float m = dec_f(amax[(long long)d * NHEAD + hd]);
  float v = __expf(a - m);
  ex[i] = v;
  atomicAdd(&denom[(long long)d * NHEAD + hd], v);
}

// ---------------- edge pass 3: out[dst] += h[src] * w  (float4 gather) ----------------
// One thread = one edge x one head x 4 consecutive channels. C is compile-time so
// all index math reduces to shifts/masks (C/4 and NHEAD*C/4 are powers of two).
template<int C>
__global__ void k_edge_scatter(const int* __restrict__ ei, long long E, long long Ep,
                               const float* __restrict__ h, const float* __restrict__ ex,
                               const float* __restrict__ denom, float* __restrict__ out) {
  constexpr int C4  = C / 4;            // float4 chunks per head
  constexpr int HC4 = NHEAD * C4;       // float4 chunks per edge
  constexpr int HC  = NHEAD * C;
  long long total = Ep * HC4;
  long long i = (long long)blockIdx.x * blockDim.x + threadIdx.x;
  if (i >= total) return;
  long long e = i / HC4;
  int q  = (int)(i - e * HC4);
  int hd = q / C4;
  int c4 = q - hd * C4;
  int s, d; get_edge(ei, E, e, s, d);
  float w = ex[e * NHEAD + hd] / (denom[(long long)d * NHEAD + hd] + 1e-16f);
  const float4 hv = *(const float4*)(h + (long long)s * HC + hd * C + c4 * 4);
  float* op = out + (long long)d * HC + hd * C + c4 * 4;
  atomicAdd(op + 0, hv.x * w);
  atomicAdd(op + 1, hv.y * w);
  atomicAdd(op + 2, hv.z * w);
  atomicAdd(op + 3, hv.w * w);
}

// ---------------- bias (+optional leaky relu) ----------------
__global__ void k_bias_act(float* __restrict__ out, const float* __restrict__ b,
                           long long n, int HC, int leaky) {
  long long i = (long long)blockIdx.x * blockDim.x + threadIdx.x;
  if (i >= n * HC) return;
  int hc = (int)(i % HC);
  float v = out[i] + b[hc];
  out[i] = leaky ? lrelu(v) : v;
}

static inline unsigned cdivll(long long a, long long b) { return (unsigned)((a + b - 1) / b); }

extern "C" void kernel_launch(void* const* d_in, const int* in_sizes, int n_in,
                              void* d_out, int out_size, void* d_ws, size_t ws_size,
                              hipStream_t stream) {
  const float* x      = (const float*)d_in[0];
  const int*   ei     = (const int*)  d_in[1];
  const float* W1     = (const float*)d_in[2];
  const float* a_src1 = (const float*)d_in[3];
  const float* a_dst1 = (const float*)d_in[4];
  const float* b1     = (const float*)d_in[5];
  const float* W2     = (const float*)d_in[6];
  const float* a_src2 = (const float*)d_in[7];
  const float* a_dst2 = (const float*)d_in[8];
  const float* b2     = (const float*)d_in[9];
  float* out = (float*)d_out;

  const long long N  = in_sizes[0] / F_IN;      // 50000
  const long long E  = in_sizes[1] / 2;         // 800000
  const long long Ep = E + N;                   // with self-loops

  // ---- workspace layout (floats) ----
  float*    h1    = (float*)d_ws;               // N*256
  float*    act1  = h1   + N * HC1;             // N*256  (layer-1 output / layer-2 input)
  float*    h2    = act1 + N * HC1;             // N*32
  float*    as_   = h2   + N * HC2;             // N*8
  float*    ad_   = as_  + N * NHEAD;           // N*8
  unsigned* amax  = (unsigned*)(ad_ + N * NHEAD); // N*8
  float*    denom = (float*)amax + N * NHEAD;   // N*8
  float*    ex    = denom + N * NHEAD;          // Ep*8

  const int B = 256;

  // =========================== Layer 1 ===========================
  k_fill_zero   <<<cdivll(N * HC1, B), B, 0, stream>>>(act1, N * HC1);
  k_init_softmax<<<cdivll(N * NHEAD, B), B, 0, stream>>>(amax, denom, N * NHEAD);

  // GEMM1: [N,128] @ [128,256] ; each wave: 16 x 128 strip (NT=8)
  {
    constexpr int NT = 8;
    int groupsN = HC1 / (16 * NT);                    // 2
    int groupsTotal = (int)(N / 16) * groupsN;        // 6250 waves
    k_gemm_wmma<NT><<<cdivll((long long)groupsTotal * 32, B), B, 0, stream>>>(
        x, W1, h1, F_IN, HC1, groupsN, groupsTotal);
  }
  k_alpha<HIDC><<<cdivll(N * NHEAD, B), B, 0, stream>>>(h1, a_src1, a_dst1, as_, ad_, N);
  k_edge_max         <<<cdivll(Ep * NHEAD, B), B, 0, stream>>>(ei, E, Ep, as_, ad_, amax);
  k_edge_exp         <<<cdivll(Ep * NHEAD, B), B, 0, stream>>>(ei, E, Ep, as_, ad_, amax, denom, ex);
  k_edge_scatter<HIDC><<<cdivll(Ep * (NHEAD * HIDC / 4), B), B, 0, stream>>>(
      ei, E, Ep, h1, ex, denom, act1);
  k_bias_act         <<<cdivll(N * HC1, B), B, 0, stream>>>(act1, b1, N, HC1, 1);

  // =========================== Layer 2 ===========================
  k_fill_zero   <<<cdivll(N * HC2, B), B, 0, stream>>>(out, N * HC2);
  k_init_softmax<<<cdivll(N * NHEAD, B), B, 0, stream>>>(amax, denom, N * NHEAD);

  // GEMM2: [N,256] @ [256,32] ; each wave: 16 x 32 strip (NT=2)
  {
    constexpr int NT = 2;
    int groupsN = HC2 / (16 * NT);                    // 1
    int groupsTotal = (int)(N / 16) * groupsN;        // 3125 waves
    k_gemm_wmma<NT><<<cdivll((long long)groupsTotal * 32, B), B, 0, stream>>>(
        act1, W2, h2, HC1, HC2, groupsN, groupsTotal);
  }
  k_alpha<OUTC><<<cdivll(N * NHEAD, B), B, 0, stream>>>(h2, a_src2, a_dst2, as_, ad_, N);
  k_edge_max         <<<cdivll(Ep * NHEAD, B), B, 0, stream>>>(ei, E, Ep, as_, ad_, amax);
  k_edge_exp         <<<cdivll(Ep * NHEAD, B), B, 0, stream>>>(ei, E, Ep, as_, ad_, amax, denom, ex);
  k_edge_scatter<OUTC><<<cdivll(Ep * (NHEAD * OUTC / 4), B), B, 0, stream>>>(
      ei, E, Ep, h2, ex, denom, out);
  k_bias_act         <<<cdivll(N * HC2, B), B, 0, stream>>>(out, b2, N, HC2, 0);
}